// MCHManagedCollisionModule_36507222016744
// MI455X (gfx1250) — compile-verified
//
#include <hip/hip_runtime.h>
#include <limits.h>

// MCH managed-collision remap: lower_bound over a 4.19M sorted table for 33.5M ids.
// Memory/latency bound: ~256MB stream (11us @ 23.3TB/s HBM floor); the 16MB int32
// table is L2-resident (192MB L2), so all probes after warm-up are L2 hits.
// Strategy: 16KB LDS pivot cache (13 halving steps at LDS latency) + 8-way
// lockstep global binary search (<=11 L2 probes, 8 loads in flight per step).

#define BLOCK 256
#define VPT   8                 // independent searches per thread (MLP)
#define VPB   (BLOCK * VPT)     // 2048 values per block
#define NPIV  4096              // 16 KB int32 pivot cache in LDS (of 320 KB/WGP)
#define PPT   (NPIV / BLOCK)    // 16 pivots staged per thread

typedef int   v4i __attribute__((ext_vector_type(4)));
typedef float v4f __attribute__((ext_vector_type(4)));

__global__ __launch_bounds__(BLOCK)
void mch_remap_kernel(const int* __restrict__ values,
                      const int* __restrict__ sorted_ids,   // ZCH_SIZE entries (search first L)
                      const int* __restrict__ mapping,      // ZCH_SIZE entries
                      const int* __restrict__ zch_index_p,  // scalar on device
                      float* __restrict__ out,
                      int n, int L)                         // L = ZCH_SIZE - 1
{
    __shared__ int pivots[NPIV];
    const int bw = (L + NPIV - 1) / NPIV;   // bucket width (1024 for ZCH=4M)

    // ---- stage pivots: 16 independent strided loads in flight, then 16 ds_stores.
    // Branch-free tail: clamp the address (always legal), cndmask the value.
    {
        int pv[PPT];
        #pragma unroll
        for (int k = 0; k < PPT; ++k) {
            int i = threadIdx.x + k * BLOCK;
            long long g  = (long long)i * bw;
            long long gc = (g < (long long)(L - 1)) ? g : (long long)(L - 1);
            int x = sorted_ids[gc];                     // unconditional load
            pv[k] = (g < (long long)L) ? x : INT_MAX;   // v_cndmask
        }
        #pragma unroll
        for (int k = 0; k < PPT; ++k)
            pivots[threadIdx.x + k * BLOCK] = pv[k];
    }
    __syncthreads();

    const float sentinel = (float)zch_index_p[0];

    const int  base = blockIdx.x * VPB + threadIdx.x * VPT;
    const bool full = (base + VPT) <= n;

    // ---- load 8 values (NT: streamed once; keep the table hot in L2) ----
    int v[VPT];
    if (full) {
        const v4i* p = (const v4i*)(values + base);     // 32B aligned (base % 8 == 0)
        v4i a = __builtin_nontemporal_load(p);
        v4i b = __builtin_nontemporal_load(p + 1);
        v[0]=a.x; v[1]=a.y; v[2]=a.z; v[3]=a.w;
        v[4]=b.x; v[5]=b.y; v[6]=b.z; v[7]=b.w;
    } else {
        #pragma unroll
        for (int j = 0; j < VPT; ++j)
            v[j] = (base + j < n) ? values[base + j] : 0;
    }

    // ---- coarse stage: cnt = #pivots strictly < v (lockstep lower_bound in LDS).
    // Split per step into "issue 8 ds_loads" then "update" so the 8 loads batch
    // under one s_wait_dscnt.
    int lo[VPT], len[VPT];
    #pragma unroll
    for (int j = 0; j < VPT; ++j) { lo[j] = 0; len[j] = NPIV; }

    #pragma unroll 1
    for (int s = 0; s < 13; ++s) {          // ceil(log2(4096+1)) = 13 halvings
        int m[VPT], x[VPT];
        #pragma unroll
        for (int j = 0; j < VPT; ++j) {
            int half = len[j] >> 1;
            m[j] = lo[j] + half;
            int mc = m[j] < (NPIV - 1) ? m[j] : (NPIV - 1);
            x[j] = pivots[mc];              // unconditional ds_load, 8 in flight
        }
        #pragma unroll
        for (int j = 0; j < VPT; ++j) {
            int half = len[j] >> 1;
            if (len[j] > 0) {               // -> v_cndmask updates
                if (x[j] < v[j]) { lo[j] = m[j] + 1; len[j] -= half + 1; }
                else             { len[j] = half; }
            }
        }
    }

    // Duplicate-safe bracket: if cnt>0 then sorted_ids[(cnt-1)*bw] < v, so the
    // answer lies in [(cnt-1)*bw, min(cnt*bw, L)]; if cnt==0 the answer is 0.
    #pragma unroll
    for (int j = 0; j < VPT; ++j) {
        int cnt = lo[j];
        int s   = (cnt > 0) ? (cnt - 1) * bw : 0;
        int e   = (cnt > 0) ? min(cnt * bw, L) : 0;
        lo[j]  = s;
        len[j] = e - s;
    }

    // ---- fine stage: lockstep lower_bound in global memory (L2 hits).
    // Index lo+half <= L is always valid (table has L+1 entries), so all 8 loads
    // issue unconditionally per step before any update consumes them.
    const int fsteps = 32 - __builtin_clz((unsigned)bw);  // 11 for bw=1024
    #pragma unroll 1
    for (int s = 0; s < fsteps; ++s) {
        int m[VPT], x[VPT];
        #pragma unroll
        for (int j = 0; j < VPT; ++j) {
            m[j] = lo[j] + (len[j] >> 1);
            x[j] = __ldg(sorted_ids + m[j]);   // 8 independent global_load_b32
        }
        #pragma unroll
        for (int j = 0; j < VPT; ++j) {
            int half = len[j] >> 1;
            if (len[j] > 0) {
                if (x[j] < v[j]) { lo[j] = m[j] + 1; len[j] -= half + 1; }
                else             { len[j] = half; }
            }
        }
    }

    // ---- final gathers + select (two more L2 hits per value, 8-way MLP) ----
    int hit[VPT], rem[VPT];
    #pragma unroll
    for (int j = 0; j < VPT; ++j) hit[j] = __ldg(sorted_ids + lo[j]);
    #pragma unroll
    for (int j = 0; j < VPT; ++j) rem[j] = __ldg(mapping   + lo[j]);

    float r[VPT];
    #pragma unroll
    for (int j = 0; j < VPT; ++j)
        r[j] = (hit[j] == v[j]) ? (float)rem[j] : sentinel;

    // ---- store (NT: output is write-once; don't pollute L2) ----
    if (full) {
        v4f* q = (v4f*)(out + base);
        v4f a, b;
        a.x=r[0]; a.y=r[1]; a.z=r[2]; a.w=r[3];
        b.x=r[4]; b.y=r[5]; b.z=r[6]; b.w=r[7];
        __builtin_nontemporal_store(a, q);
        __builtin_nontemporal_store(b, q + 1);
    } else {
        #pragma unroll
        for (int j = 0; j < VPT; ++j)
            if (base + j < n) out[base + j] = r[j];
    }
}

extern "C" void kernel_launch(void* const* d_in, const int* in_sizes, int n_in,
                              void* d_out, int out_size, void* d_ws, size_t ws_size,
                              hipStream_t stream) {
    const int* values     = (const int*)d_in[0];
    const int* sorted_ids = (const int*)d_in[1];
    const int* mapping    = (const int*)d_in[2];
    const int* zch_index  = (const int*)d_in[3];
    float*     out        = (float*)d_out;

    const int n = in_sizes[0];
    const int L = in_sizes[1] - 1;          // search over table[:-1] (matches reference)

    const int grid = (n + VPB - 1) / VPB;
    mch_remap_kernel<<<grid, BLOCK, 0, stream>>>(values, sorted_ids, mapping,
                                                 zch_index, out, n, L);
}